// GO_sim_embedding_66185446032026
// MI455X (gfx1250) — compile-verified
//
#include <hip/hip_runtime.h>

// ---------------------------------------------------------------------------
// GCN triple-branch: agg = segment_sum(x[src], dst);  out = relu(agg@W+b) + x
// d_out (3 * 50000 * 128 f32) doubles as the agg accumulator (in-place GEMM).
// CDNA5 paths: v_wmma_f32_16x16x4_f32 (exact fp32 GEMM),
//              global_load_async_to_lds_b128 + s_wait_asynccnt (A-tile stage),
//              global_atomic_add_f32 (scatter phase).
// ---------------------------------------------------------------------------

#define FEAT 128

typedef float v2f __attribute__((ext_vector_type(2)));
typedef float v8f __attribute__((ext_vector_type(8)));

// ---- zero the accumulator (d_out is poisoned by the harness) ----
__global__ void go_zero_f4(float4* __restrict__ p, long n4) {
  long i = (long)blockIdx.x * blockDim.x + threadIdx.x;
  if (i < n4) p[i] = make_float4(0.f, 0.f, 0.f, 0.f);
}

// ---- one wave per edge: gather 512B row of x[src], atomic-add into agg[dst]
__global__ void go_scatter_add(const float* __restrict__ x,
                               const int* __restrict__ src,
                               const int* __restrict__ dst,
                               float* __restrict__ agg,
                               int n_edges) {
  long tid = (long)blockIdx.x * blockDim.x + threadIdx.x;
  long e = tid >> 5;                 // edge id: uniform across the wave32
  if (e >= n_edges) return;
  int c = (int)(tid & 31) * 4;       // float4 chunk within the 128-float row
  long s = src[e];
  long d = dst[e];
  const float4 v = *reinterpret_cast<const float4*>(x + s * FEAT + c);
  float* p = agg + d * FEAT + c;
  unsafeAtomicAdd(p + 0, v.x);       // native global_atomic_add_f32
  unsafeAtomicAdd(p + 1, v.y);
  unsafeAtomicAdd(p + 2, v.z);
  unsafeAtomicAdd(p + 3, v.w);
}

// ---- per-16-row tile: out = relu(agg @ W + b) + x, in place over agg ----
// Block = 256 threads = 8 waves; wave w owns output columns [16w, 16w+16).
// A tile staged in LDS via async copy (padded stride 132 -> conflict-free
// ds_load_2addr_b64 fragment reads).
#define A_STRIDE 132

__global__ __launch_bounds__(256) void go_gcn_tile(
    float* __restrict__ inout,        // agg in, result out (same rows only)
    const float* __restrict__ x,      // residual input
    const float* __restrict__ W,      // 128x128 row-major (k, n)
    const float* __restrict__ bias,   // 128
    int n_nodes) {
  __shared__ float ldsA[16 * A_STRIDE];
  const int tile = blockIdx.x;
  const int t = threadIdx.x;

  // Stage the 16x128 A tile: thread t owns 8 contiguous floats of one row.
  // Hot path: two async b128 copies straight into LDS (no VGPR round-trip);
  // IOFFSET is added to both the global and LDS addresses (ISA 08, GV mode).
  {
    const int row = t >> 4;                 // 0..15
    const int col = (t & 15) * 8;           // 0..120
    const long grow = (long)tile * 16 + row;
    float* lp = ldsA + row * A_STRIDE + col;
    if (grow < n_nodes) {
      const unsigned lds_off = (unsigned)(size_t)lp;           // LDS byte addr
      const float* gp = inout + grow * FEAT + col;
      asm volatile(
          "global_load_async_to_lds_b128 %0, %1, off\n\t"
          "global_load_async_to_lds_b128 %0, %1, off offset:16"
          :: "v"(lds_off), "v"(gp) : "memory");
    } else {
      // cold guard path (never taken for n_nodes % 16 == 0): zero-fill
      float4* l = reinterpret_cast<float4*>(lp);
      l[0] = make_float4(0.f, 0.f, 0.f, 0.f);
      l[1] = make_float4(0.f, 0.f, 0.f, 0.f);
    }
  }
  asm volatile("s_wait_asynccnt 0x0" ::: "memory");
  __syncthreads();

  const int lane = t & 31;
  const int half = lane >> 4;          // lane group 0/1 (selects K pair)
  const int l16  = lane & 15;
  const int nb   = (t >> 5) * 16;      // wave -> N tile base
  const int n    = nb + l16;

  // D = A(16x128) x W(128x[nb..nb+16)) accumulated over 32 K=4 WMMA steps.
  v8f acc = {};
#pragma unroll
  for (int kk = 0; kk < 32; ++kk) {
    const int k0 = kk * 4 + half * 2;
    // A fragment: lane l -> A[l&15][k0], A[l&15][k0+1]  (float2 LDS read)
    const float2 af = *reinterpret_cast<const float2*>(ldsA + l16 * A_STRIDE + k0);
    v2f a; a.x = af.x; a.y = af.y;
    // B fragment: lane l -> W[k0][n], W[k0+1][n]
    v2f b; b.x = W[k0 * FEAT + n]; b.y = W[(k0 + 1) * FEAT + n];
    acc = __builtin_amdgcn_wmma_f32_16x16x4_f32(false, a, false, b,
                                                (short)0, acc, false, false);
  }

  // Epilogue: C/D VGPR r holds row (r + 8*half), col n.
  const float bv = bias[n];
#pragma unroll
  for (int r = 0; r < 8; ++r) {
    const int m = r + half * 8;
    const long grow = (long)tile * 16 + m;
    if (grow < n_nodes) {
      float v = acc[r] + bv;
      v = v > 0.f ? v : 0.f;
      v += x[grow * FEAT + n];
      inout[grow * FEAT + n] = v;   // only rows of this tile; staged pre-barrier
    }
  }
}

extern "C" void kernel_launch(void* const* d_in, const int* in_sizes, int n_in,
                              void* d_out, int out_size, void* d_ws, size_t ws_size,
                              hipStream_t stream) {
  (void)n_in; (void)d_ws; (void)ws_size;
  const float* x[3]   = {(const float*)d_in[0], (const float*)d_in[1], (const float*)d_in[2]};
  const int*  src[3]  = {(const int*)d_in[3], (const int*)d_in[5], (const int*)d_in[7]};
  const int*  dst[3]  = {(const int*)d_in[4], (const int*)d_in[6], (const int*)d_in[8]};
  const float* W[3]   = {(const float*)d_in[9],  (const float*)d_in[10], (const float*)d_in[11]};
  const float* bia[3] = {(const float*)d_in[12], (const float*)d_in[13], (const float*)d_in[14]};
  float* out = (float*)d_out;

  const int  n_nodes = in_sizes[0] / FEAT;           // 50000
  const long per     = (long)n_nodes * FEAT;         // floats per branch output

  // 1) zero the three agg buffers (= d_out)
  const long n4 = (long)out_size / 4;
  go_zero_f4<<<(int)((n4 + 255) / 256), 256, 0, stream>>>((float4*)d_out, n4);

  // 2) scatter-add per branch (one wave32 per edge)
  for (int br = 0; br < 3; ++br) {
    const int  n_edges = in_sizes[3 + 2 * br];
    const long threads = (long)n_edges * 32;
    go_scatter_add<<<(int)((threads + 255) / 256), 256, 0, stream>>>(
        x[br], src[br], dst[br], out + (long)br * per, n_edges);
  }

  // 3) fp32 WMMA GEMM + relu + bias + residual, in place over agg
  const int n_tiles = (n_nodes + 15) / 16;           // 3125
  for (int br = 0; br < 3; ++br) {
    go_gcn_tile<<<n_tiles, 256, 0, stream>>>(
        out + (long)br * per, x[br], W[br], bia[br], n_nodes);
  }
}